// SCoNeLayer_62886911148584
// MI455X (gfx1250) — compile-verified
//
#include <hip/hip_runtime.h>

#define CC 128   // channel dim, fixed by reference

typedef float v2f __attribute__((ext_vector_type(2)));
typedef float v8f __attribute__((ext_vector_type(8)));

// ---------------------------------------------------------------------------
// K0: zero node_vals accumulator in workspace
// ---------------------------------------------------------------------------
__global__ __launch_bounds__(256) void scone_zero(float* __restrict__ p, int n4) {
    int i = blockIdx.x * 256 + threadIdx.x;
    if (i < n4) ((float4*)p)[i] = make_float4(0.f, 0.f, 0.f, 0.f);
}

// ---------------------------------------------------------------------------
// K1: fused triple GEMM  y0=x@w0, z_id=x@w1, y2=x@w2  via V_WMMA_F32_16X16X4_F32
//     - x (A operand) loaded ONCE and reused for all three weights
//     - z_id written directly to d_out (accumulation base)
//     - y2 written to workspace (needed for triangle gathers)
//     - y0 scattered (signed) into node_vals with fp32 atomics
// Wave tile: 64 rows (4 m-tiles) x 32 cols (2 n-tiles), 3 weights
//   -> acc[3][4][2] v8f = 192 VGPRs, 24 WMMA per K-step of 4.
// Block = 256 threads (8 waves) covers 128 rows x full 128 cols.
// ---------------------------------------------------------------------------
__global__ __launch_bounds__(256) void scone_gemm3(
    const float* __restrict__ x,
    const float* __restrict__ w0,
    const float* __restrict__ w1,
    const float* __restrict__ w2,
    const int*   __restrict__ edge_nodes,   // [E,2] (tail, head)
    float* __restrict__ z_out,              // d_out: receives z_id
    float* __restrict__ y2_out,             // ws
    float* __restrict__ node_vals,          // ws, pre-zeroed
    int n_edges)
{
    const int lane  = threadIdx.x & 31;
    const int wid   = threadIdx.x >> 5;
    const int nbase = (wid & 3) * 32;                       // 0,32,64,96
    const int mbase = blockIdx.x * 128 + (wid >> 2) * 64;   // row base
    if (mbase >= n_edges) return;

    const int l15   = lane & 15;
    const int khalf = (lane >> 4) * 2;   // 0 for lanes 0-15, 2 for lanes 16-31
    const int hi8   = (lane >> 4) * 8;   // C/D row offset for high half-wave

    const float* wptr[3] = { w0, w1, w2 };

    v8f acc[3][4][2];
    #pragma unroll
    for (int w = 0; w < 3; ++w)
        #pragma unroll
        for (int mt = 0; mt < 4; ++mt)
            #pragma unroll
            for (int nt = 0; nt < 2; ++nt)
                acc[w][mt][nt] = (v8f){0.f,0.f,0.f,0.f,0.f,0.f,0.f,0.f};

    for (int k = 0; k < CC; k += 4) {
        // A tiles (x): ISA A-layout 16x4 f32 -> lane holds x[m][k+khalf .. +1] (8B load)
        v2f A[4];
        #pragma unroll
        for (int mt = 0; mt < 4; ++mt) {
            const float* ap = x + (size_t)(mbase + mt * 16 + l15) * CC + k + khalf;
            A[mt] = *(const v2f*)ap;
        }
        #pragma unroll
        for (int w = 0; w < 3; ++w) {
            #pragma unroll
            for (int nt = 0; nt < 2; ++nt) {
                // B tile (w): lane holds w[k+khalf][n], w[k+khalf+1][n]
                const float* bp = wptr[w] + (size_t)(k + khalf) * CC + nbase + nt * 16 + l15;
                v2f B;
                B.x = bp[0];
                B.y = bp[CC];
                #pragma unroll
                for (int mt = 0; mt < 4; ++mt) {
                    acc[w][mt][nt] = __builtin_amdgcn_wmma_f32_16x16x4_f32(
                        false, A[mt], false, B, (short)0, acc[w][mt][nt], false, false);
                }
            }
        }
    }

    // ---- epilogue: z_id -> d_out, y2 -> ws (C/D layout: reg r -> rows r / r+8)
    #pragma unroll
    for (int mt = 0; mt < 4; ++mt) {
        #pragma unroll
        for (int r = 0; r < 8; ++r) {
            const size_t row = (size_t)(mbase + mt * 16 + r + hi8);
            const int colb = nbase + l15;
            #pragma unroll
            for (int nt = 0; nt < 2; ++nt) {
                z_out [row * CC + colb + nt * 16] = acc[1][mt][nt][r];
                y2_out[row * CC + colb + nt * 16] = acc[2][mt][nt][r];
            }
        }
    }

    // ---- scatter y0 with boundary signs s1 = (-1 tail, +1 head) into node_vals
    #pragma unroll
    for (int mt = 0; mt < 4; ++mt) {
        #pragma unroll
        for (int r = 0; r < 8; ++r) {
            const int e    = mbase + mt * 16 + r + hi8;
            const int tail = edge_nodes[2 * e];
            const int head = edge_nodes[2 * e + 1];
            const int colb = nbase + l15;
            #pragma unroll
            for (int nt = 0; nt < 2; ++nt) {
                const float v = acc[0][mt][nt][r];
                atomicAdd(node_vals + (size_t)head * CC + colb + nt * 16,  v);
                atomicAdd(node_vals + (size_t)tail * CC + colb + nt * 16, -v);
            }
        }
    }
}

// ---------------------------------------------------------------------------
// K2: per-triangle signed gather of y2 rows, signed scatter-add into d_out.
// 32 threads per triangle, float4 per thread.
// ---------------------------------------------------------------------------
__global__ __launch_bounds__(256) void scone_tri(
    const float* __restrict__ y2,
    const int*   __restrict__ tri_edges,   // [T,3]
    float* __restrict__ z_out,
    int n_tris)
{
    const int idx = blockIdx.x * 256 + threadIdx.x;
    const int t = idx >> 5;
    if (t >= n_tris) return;
    const int c4 = (idx & 31) << 2;
    const int e0 = tri_edges[3 * t];
    const int e1 = tri_edges[3 * t + 1];
    const int e2 = tri_edges[3 * t + 2];
    const float4 a = *(const float4*)(y2 + (size_t)e0 * CC + c4);
    const float4 b = *(const float4*)(y2 + (size_t)e1 * CC + c4);
    const float4 c = *(const float4*)(y2 + (size_t)e2 * CC + c4);
    const float tv[4] = { a.x - b.x + c.x, a.y - b.y + c.y,
                          a.z - b.z + c.z, a.w - b.w + c.w };
    #pragma unroll
    for (int j = 0; j < 4; ++j) {
        atomicAdd(z_out + (size_t)e0 * CC + c4 + j,  tv[j]);   // s2 = +1
        atomicAdd(z_out + (size_t)e1 * CC + c4 + j, -tv[j]);   // s2 = -1
        atomicAdd(z_out + (size_t)e2 * CC + c4 + j,  tv[j]);   // s2 = +1
    }
}

// ---------------------------------------------------------------------------
// K3: z_lower gather + tanh epilogue. 32 threads/edge, float4 each.
// ---------------------------------------------------------------------------
__global__ __launch_bounds__(256) void scone_finish(
    const float* __restrict__ node_vals,
    const int*   __restrict__ edge_nodes,
    float* __restrict__ z_out,
    int n_edges)
{
    const int idx = blockIdx.x * 256 + threadIdx.x;
    const int e = idx >> 5;
    if (e >= n_edges) return;
    const int c4 = (idx & 31) << 2;
    const int tail = edge_nodes[2 * e];
    const int head = edge_nodes[2 * e + 1];
    const float4 h = *(const float4*)(node_vals + (size_t)head * CC + c4);
    const float4 t = *(const float4*)(node_vals + (size_t)tail * CC + c4);
    float4 z = *(float4*)(z_out + (size_t)e * CC + c4);
    z.x = tanhf(z.x + h.x - t.x);
    z.y = tanhf(z.y + h.y - t.y);
    z.z = tanhf(z.z + h.z - t.z);
    z.w = tanhf(z.w + h.w - t.w);
    *(float4*)(z_out + (size_t)e * CC + c4) = z;
}

// ---------------------------------------------------------------------------
extern "C" void kernel_launch(void* const* d_in, const int* in_sizes, int n_in,
                              void* d_out, int out_size, void* d_ws, size_t ws_size,
                              hipStream_t stream) {
    const float* x  = (const float*)d_in[0];
    const float* w0 = (const float*)d_in[1];
    const float* w1 = (const float*)d_in[2];
    const float* w2 = (const float*)d_in[3];
    const int* edge_nodes = (const int*)d_in[4];
    const int* tri_edges  = (const int*)d_in[5];

    const int E = in_sizes[4] / 2;     // 400000
    const int T = in_sizes[5] / 3;     // 150000
    const int N_NODES = 100000;        // fixed by reference setup

    float* y2        = (float*)d_ws;                       // E*CC floats
    float* node_vals = (float*)d_ws + (size_t)E * CC;      // N_NODES*CC floats
    float* out       = (float*)d_out;

    const int nzero4 = N_NODES * CC / 4;
    scone_zero<<<(nzero4 + 255) / 256, 256, 0, stream>>>(node_vals, nzero4);

    scone_gemm3<<<(E + 127) / 128, 256, 0, stream>>>(
        x, w0, w1, w2, edge_nodes, out, y2, node_vals, E);

    const long long ttot = (long long)T * 32;
    scone_tri<<<(unsigned)((ttot + 255) / 256), 256, 0, stream>>>(
        y2, tri_edges, out, T);

    const long long etot = (long long)E * 32;
    scone_finish<<<(unsigned)((etot + 255) / 256), 256, 0, stream>>>(
        node_vals, edge_nodes, out, E);
}